// ECE_18631568130668
// MI455X (gfx1250) — compile-verified
//
#include <hip/hip_runtime.h>
#include <hip/hip_bf16.h>

#define NBINS 10
#define BLK   256
#define NB1   2048   // pass-1 blocks (multiple of 4 for WMMA K-chunks)

typedef __attribute__((ext_vector_type(2))) float v2f;
typedef __attribute__((ext_vector_type(8))) float v8f;

// ---------------------------------------------------------------------------
// Pass 1: streaming histogram. Thread-private LDS columns [bin][tid] -> zero
// bank conflicts (bank = tid & 63, 32 distinct banks per wave). cnt/acc packed
// into one u32 (block totals <= 256*64 = 16384 < 2^16 -> no carry between
// halves). Partials written as rows: row r = stat*10 + bin, column = block.
// Rows 30,31 zero-padded so pass 2 WMMA can consume a clean 32-row matrix.
// ---------------------------------------------------------------------------
__global__ void ece_pass1(const float* __restrict__ conf,
                          const int*  __restrict__ pred,
                          const int*  __restrict__ lab,
                          float* __restrict__ partials,
                          int n, int nb1) {
    __shared__ float    s_conf[NBINS * BLK];
    __shared__ unsigned s_pack[NBINS * BLK];

    const int tid = threadIdx.x;
    const int gid = blockIdx.x * BLK + tid;
    const int gthreads = gridDim.x * BLK;

    #pragma unroll
    for (int b = 0; b < NBINS; ++b) {
        s_conf[b * BLK + tid] = 0.0f;
        s_pack[b * BLK + tid] = 0u;
    }
    // no barrier needed: each thread touches only its own column until reduce

    const int n4 = n >> 2;
    const float4* __restrict__ c4 = (const float4*)conf;
    const int4*   __restrict__ p4 = (const int4*)pred;
    const int4*   __restrict__ l4 = (const int4*)lab;

    for (int i = gid; i < n4; i += gthreads) {
        float4 c = c4[i];
        int4   p = p4[i];
        int4   l = l4[i];
        float cv[4] = {c.x, c.y, c.z, c.w};
        int   pv[4] = {p.x, p.y, p.z, p.w};
        int   lv[4] = {l.x, l.y, l.z, l.w};
        #pragma unroll
        for (int j = 0; j < 4; ++j) {
            int b = (int)ceilf(cv[j] * 10.0f) - 1;
            b = b < 0 ? 0 : (b > NBINS - 1 ? NBINS - 1 : b);
            unsigned inc = (pv[j] == lv[j]) ? 0x10001u : 1u;
            atomicAdd(&s_conf[b * BLK + tid], cv[j]);   // ds_add_f32, private slot
            atomicAdd(&s_pack[b * BLK + tid], inc);     // ds_add_u32, private slot
        }
    }
    // scalar tail (none for N = 2^25, kept for generality)
    for (int i = (n4 << 2) + gid; i < n; i += gthreads) {
        float cvs = conf[i];
        int b = (int)ceilf(cvs * 10.0f) - 1;
        b = b < 0 ? 0 : (b > NBINS - 1 ? NBINS - 1 : b);
        unsigned inc = (pred[i] == lab[i]) ? 0x10001u : 1u;
        atomicAdd(&s_conf[b * BLK + tid], cvs);
        atomicAdd(&s_pack[b * BLK + tid], inc);
    }

    __syncthreads();

    // Block reduce: wave w handles bins {w, w+8}; conflict-free LDS reads.
    const int lane = tid & 31;
    const int w    = tid >> 5;
    for (int b = w; b < NBINS; b += 8) {
        float    cs = 0.0f;
        unsigned pk = 0u;
        #pragma unroll
        for (int t = 0; t < BLK; t += 32) {
            cs += s_conf[b * BLK + lane + t];
            pk += s_pack[b * BLK + lane + t];
        }
        #pragma unroll
        for (int off = 16; off > 0; off >>= 1) {
            cs += __shfl_down(cs, off, 32);
            pk += __shfl_down(pk, off, 32);
        }
        if (lane == 0) {
            partials[(size_t)(0 * NBINS + b) * nb1 + blockIdx.x] = cs;                   // conf sum
            partials[(size_t)(1 * NBINS + b) * nb1 + blockIdx.x] = (float)(pk & 0xFFFFu); // count
            partials[(size_t)(2 * NBINS + b) * nb1 + blockIdx.x] = (float)(pk >> 16);     // correct
        }
    }
    if (tid == 0) {  // pad rows 30,31 so the WMMA A-matrix reads zeros there
        partials[(size_t)30 * nb1 + blockIdx.x] = 0.0f;
        partials[(size_t)31 * nb1 + blockIdx.x] = 0.0f;
    }
}

// ---------------------------------------------------------------------------
// Pass 2: one wave32. Row-sum the 32 x nb1 partial matrix with
// V_WMMA_F32_16X16X4_F32: D = A x ones + C. A (16x4 f32) per ISA layout:
// lanes 0-15 hold K=0,1 in VGPR0,1; lanes 16-31 hold K=2,3. B = all-ones is
// layout-independent. Two accumulator chains cover rows 0-15 and 16-31.
// Counts stay < 2^24 -> f32 accumulation exact.
// ---------------------------------------------------------------------------
__global__ void ece_pass2(const float* __restrict__ partials,
                          float* __restrict__ out, int nb1, int n) {
    __shared__ float rs[32];

#if defined(__gfx1250__) && __has_builtin(__builtin_amdgcn_wmma_f32_16x16x4_f32)
    const int lane = threadIdx.x;
    const int rowA = lane & 15;
    const int kh   = (lane >> 4) << 1;  // 0 for lanes 0-15, 2 for lanes 16-31
    const float* base0 = partials + (size_t)rowA * nb1 + kh;
    const float* base1 = partials + (size_t)(rowA + 16) * nb1 + kh;

    v2f bones = {1.0f, 1.0f};
    v8f c0 = {};
    v8f c1 = {};
    for (int k = 0; k < nb1; k += 4) {
        v2f a0 = *(const v2f*)(base0 + k);
        v2f a1 = *(const v2f*)(base1 + k);
        c0 = __builtin_amdgcn_wmma_f32_16x16x4_f32(false, a0, false, bones,
                                                   (short)0, c0, false, false);
        c1 = __builtin_amdgcn_wmma_f32_16x16x4_f32(false, a1, false, bones,
                                                   (short)0, c1, false, false);
    }
    // D layout: VGPR v, lanes 0-15 -> row v; lanes 16-31 -> row v+8.
    if (lane == 0) {
        #pragma unroll
        for (int v = 0; v < 8; ++v) { rs[v] = c0[v]; rs[16 + v] = c1[v]; }
    }
    if (lane == 16) {
        #pragma unroll
        for (int v = 0; v < 8; ++v) { rs[8 + v] = c0[v]; rs[24 + v] = c1[v]; }
    }
#else
    const int lane = threadIdx.x;
    float s = 0.0f;
    if (lane < 30) {
        const float* row = partials + (size_t)lane * nb1;
        for (int k = 0; k < nb1; ++k) s += row[k];
    }
    rs[lane] = s;
#endif

    __syncthreads();
    if (threadIdx.x == 0) {
        float ece = 0.0f;
        float fn = (float)n;
        #pragma unroll
        for (int b = 0; b < NBINS; ++b) {
            float cs = rs[b];
            float cn = rs[NBINS + b];
            float ac = rs[2 * NBINS + b];
            if (cn > 0.0f) {
                float avg_conf = cs / cn;
                float avg_acc  = ac / cn;
                ece += fabsf(avg_conf - avg_acc) * (cn / fn);
            }
        }
        out[0] = ece;
    }
}

extern "C" void kernel_launch(void* const* d_in, const int* in_sizes, int n_in,
                              void* d_out, int out_size, void* d_ws, size_t ws_size,
                              hipStream_t stream) {
    const float* conf = (const float*)d_in[0];
    const int*   pred = (const int*)d_in[1];
    const int*   lab  = (const int*)d_in[2];
    const int    n    = in_sizes[0];
    float* out      = (float*)d_out;
    float* partials = (float*)d_ws;

    int nb1 = NB1;
    size_t need = (size_t)32 * (size_t)nb1 * sizeof(float);
    if (ws_size < need) {  // shrink deterministically if scratch is small
        nb1 = (int)(ws_size / (32 * sizeof(float))) & ~3;
        if (nb1 < 4) nb1 = 4;
    }

    ece_pass1<<<nb1, BLK, 0, stream>>>(conf, pred, lab, partials, n, nb1);
    ece_pass2<<<1, 32, 0, stream>>>(partials, out, nb1, n);
}